// NonlinearLayer_69114613727724
// MI455X (gfx1250) — compile-verified
//
#include <hip/hip_runtime.h>
#include <stdint.h>

// ---------------------------------------------------------------------------
// Per-channel piecewise-linear calibration of (real, imag) -> complex64 out.
//
// Roofline: 256 MB read + 256 MB write @ 23.3 TB/s ~= 22 us floor. fp32 is
// fixed by the reference and there is no contraction => WMMA inapplicable;
// the chip-specific wins are the async global->LDS table DMA (ASYNCcnt),
// non-temporal b128 streaming (512 MB > 192 MB L2), and a bank-conflict-free
// Eytzinger probe tree in LDS:
//   - rank search over 101 breakpoints padded to a perfect 127-node BFS tree
//   - top 3 levels in registers (6 uniform VGPRs + literal +INF)
//   - 4 LDS levels probe CONTIGUOUS words per level -> 0 bank conflicts
//   - one {slope, intercept} gather + fma per value (no per-element divide)
// LDS traffic: 6 conflict-free b32 per value = 24B LDS per 8B HBM, safely
// under the ~3.5x LDS:HBM bandwidth ratio of a WGP machine.
// ---------------------------------------------------------------------------

#define NUM_BP   101                 // breakpoints per channel (K)
#define TREE_N   127                 // padded perfect tree (7 levels)
#define THREADS  256                 // 8 wave32 waves per block
#define EPT      8                   // elements per thread (per input tensor)
#define EPB      (THREADS * EPT)     // 2048 elements per block
#define HW       65536               // 256*256 plane size (fixed by reference)
#define BLOCKS_PER_PLANE (HW / EPB)  // 32

typedef float f32x4 __attribute__((ext_vector_type(4)));

__global__ __launch_bounds__(THREADS)
void pwl_complex_kernel(const float* __restrict__ xr,
                        const float* __restrict__ xi,
                        const float* __restrict__ xp,
                        const float* __restrict__ yp,
                        float* __restrict__ out,
                        int c_mask)          // num_channels - 1 (pow2)
{
    __shared__ float s_xp[NUM_BP];           // sorted breakpoints (staging)
    __shared__ float s_yp[NUM_BP];           // y values (staging)
    __shared__ float s_tree[TREE_N];         // Eytzinger probe tree
    __shared__ float s_slope[NUM_BP - 1];    // per-segment slope
    __shared__ float s_icpt[NUM_BP - 1];     // per-segment intercept

    const int plane = blockIdx.x / BLOCKS_PER_PLANE;      // b * C + c
    const int chunk = blockIdx.x - plane * BLOCKS_PER_PLANE;
    const int c     = plane & c_mask;

    // ---- Stage this channel's table into LDS via CDNA5 async global->LDS ---
    {
        const int t = threadIdx.x;
        if (t < NUM_BP) {
            const float* gx = xp + (size_t)c * NUM_BP + t;
            const float* gy = yp + (size_t)c * NUM_BP + t;
            // LDS byte address = low 32 bits of the generic shared pointer
            unsigned lx = (unsigned)(uintptr_t)&s_xp[t];
            unsigned ly = (unsigned)(uintptr_t)&s_yp[t];
            asm volatile("global_load_async_to_lds_b32 %0, %1, off"
                         :: "v"(lx), "v"(gx) : "memory");
            asm volatile("global_load_async_to_lds_b32 %0, %1, off"
                         :: "v"(ly), "v"(gy) : "memory");
        }
        asm volatile("s_wait_asynccnt 0" ::: "memory");
    }
    __syncthreads();

    // Build per-block tables once:
    //  * {slope, intercept}: y = fma(x, slope, icpt) -> no per-element divide
    //  * Eytzinger tree over 127 slots; slots >= 101 hold +INF
    if (threadIdx.x < NUM_BP - 1) {
        const float x0 = s_xp[threadIdx.x], x1 = s_xp[threadIdx.x + 1];
        const float y0 = s_yp[threadIdx.x], y1 = s_yp[threadIdx.x + 1];
        const float sl = (y1 - y0) / (x1 - x0);
        s_slope[threadIdx.x] = sl;
        s_icpt[threadIdx.x]  = fmaf(-sl, x0, y0);   // y0 - sl*x0
    }
    if (threadIdx.x < TREE_N) {
        const int h = threadIdx.x + 1;              // 1-based heap index
        const int d = 31 - __clz(h);                // depth 0..6
        const int s = ((h - (1 << d)) << (7 - d)) + (1 << (6 - d)) - 1;
        s_tree[threadIdx.x] = (s < NUM_BP) ? s_xp[s] : INFINITY;
    }
    __syncthreads();

    // Top 3 tree levels held in registers (uniform per block):
    // tree[0]=xp[63], tree[1]=xp[31], tree[2]=xp[95],
    // tree[3]=xp[15], tree[4]=xp[47], tree[5]=xp[79], tree[6]=+INF.
    const float t0 = s_xp[63];
    const float t1 = s_xp[31], t2 = s_xp[95];
    const float t3 = s_xp[15], t4 = s_xp[47], t5 = s_xp[79];

    // ---- Streaming compute: 8 real + 8 imag elements per thread ----
    const size_t base = (size_t)plane * HW + (size_t)chunk * EPB
                      + (size_t)threadIdx.x * EPT;

    const f32x4 r0 = __builtin_nontemporal_load((const f32x4*)(xr + base));
    const f32x4 r1 = __builtin_nontemporal_load((const f32x4*)(xr + base + 4));
    const f32x4 i0 = __builtin_nontemporal_load((const f32x4*)(xi + base));
    const f32x4 i1 = __builtin_nontemporal_load((const f32x4*)(xi + base + 4));

    // Interleaved (real, imag) so outputs form contiguous complex pairs.
    float v[16] = { r0.x, i0.x, r0.y, i0.y, r0.z, i0.z, r0.w, i0.w,
                    r1.x, i1.x, r1.y, i1.y, r1.z, i1.z, r1.w, i1.w };

    // Register levels 0..2 of the Eytzinger walk: k = 2k+1+(tree[k]<=x).
    // After 3 levels k in [7,14]; kb = 4*k.
    unsigned kb[16];
#pragma unroll
    for (int k = 0; k < 16; ++k) {
        const float x  = v[k];
        const bool  c0 = (t0 <= x);
        const float b1 = c0 ? t2 : t1;
        const bool  c1 = (b1 <= x);
        const float b2 = c0 ? (c1 ? INFINITY : t5) : (c1 ? t4 : t3);
        const bool  c2 = (b2 <= x);
        unsigned pb = 28u;                          // 4*7
        pb += c0 ? 16u : 0u;
        pb += c1 ? 8u  : 0u;
        pb += c2 ? 4u  : 0u;
        kb[k] = pb;
    }
    // LDS levels 3..6: each level's probes are CONTIGUOUS words in BFS order
    // (words 7..14, 15..30, 31..62, 63..126) -> conflict-free on 64 banks.
#pragma unroll
    for (int lvl = 0; lvl < 4; ++lvl) {
#pragma unroll
        for (int k = 0; k < 16; ++k) {
            const float bp = *(const float*)((const char*)s_tree + kb[k]);
            kb[k] = (kb[k] << 1) + ((bp <= v[k]) ? 8u : 4u);
        }
    }

    float o[16];
#pragma unroll
    for (int k = 0; k < 16; ++k) {
        // rank = kb/4 - 127; segment byte offset = 4*clamp(rank-1, 0, 99)
        int jb = (int)kb[k] - 512;                  // 4*(rank-1)
        jb = jb < 0 ? 0 : (jb > 4 * (NUM_BP - 2) ? 4 * (NUM_BP - 2) : jb);
        const float sl = *(const float*)((const char*)s_slope + jb);
        const float ic = *(const float*)((const char*)s_icpt  + jb);
        o[k] = fmaf(v[k], sl, ic);
    }

    // 4x b128 non-temporal stores of interleaved complex output.
    f32x4* o4 = (f32x4*)(out + 2 * base);
#pragma unroll
    for (int q = 0; q < 4; ++q) {
        f32x4 w = { o[4*q + 0], o[4*q + 1], o[4*q + 2], o[4*q + 3] };
        __builtin_nontemporal_store(w, o4 + q);
    }
}

extern "C" void kernel_launch(void* const* d_in, const int* in_sizes, int n_in,
                              void* d_out, int out_size, void* d_ws, size_t ws_size,
                              hipStream_t stream)
{
    const float* xr = (const float*)d_in[0];
    const float* xi = (const float*)d_in[1];
    const float* xp = (const float*)d_in[2];
    const float* yp = (const float*)d_in[3];
    float* out = (float*)d_out;

    const int total        = in_sizes[0];            // B*C*H*W
    const int planes       = total / HW;             // B*C
    const int num_channels = in_sizes[2] / NUM_BP;   // C (=128, power of two)

    dim3 grid(planes * BLOCKS_PER_PLANE), block(THREADS);
    pwl_complex_kernel<<<grid, block, 0, stream>>>(xr, xi, xp, yp, out,
                                                   num_channels - 1);
}